// GNNLayer_52029233824474
// MI455X (gfx1250) — compile-verified
//
#include <hip/hip_runtime.h>
#include <cmath>

// ---------------- problem constants ----------------
#define NN    50000
#define INF_  256
#define WSZ   128
#define OUTF  32
#define HH    4
#define DD    128      // OUTF * HH
#define EE    800000
#define MM    2
#define TOPK  10
#define NEG_SLOPE 0.2f

typedef __attribute__((ext_vector_type(2))) float v2f;
typedef __attribute__((ext_vector_type(8))) float v8f;

// =====================================================================
// FP32 WMMA GEMM:  C[n][j] = sum_k A[n][k] * W[j][k]
//   A: nrows x Kdim (row major), W: 128 x Kdim (row major), C: nrows x 128
// One wave owns a 16-row x 128-col strip (8 v8f accumulators).
// Weight panel staged through LDS in K-chunks of 64.
// =====================================================================
#define KCHUNK      64
#define LDSB_STRIDE 66   // 64 + pad (even -> 8B-aligned float2 reads)

__global__ __launch_bounds__(256)
void gemm_xwt_kernel(const float* __restrict__ A, const float* __restrict__ W,
                     float* __restrict__ C, int nrows, int Kdim) {
  __shared__ float ldsW[128 * LDSB_STRIDE];   // ~33 KB

  const int t     = threadIdx.x;
  const int wave  = t >> 5;
  const int lane  = t & 31;
  const int lrow  = lane & 15;           // A row / B col within the 16-tile
  const int khalf = (lane >> 4) << 1;    // 0 or 2 (K sub-offset per ISA layout)

  const int ntiles = nrows >> 4;         // 50000/16 = 3125 exact
  int tile = blockIdx.x * 8 + wave;
  const bool active = (tile < ntiles);
  if (!active) tile = ntiles - 1;        // clamp: keep control flow uniform
  const int rowBase = tile << 4;
  const float* Arow = A + (size_t)(rowBase + lrow) * Kdim;

  v8f acc[8] = {};

  const int nchunks = Kdim / KCHUNK;
  for (int ch = 0; ch < nchunks; ++ch) {
    const int k0 = ch * KCHUNK;

    // cooperative load of W[:, k0:k0+64] -> LDS (4096 float2 / 256 threads)
    #pragma unroll
    for (int it = 0; it < 16; ++it) {
      const int i   = t + (it << 8);     // 0..4095
      const int j   = i >> 5;            // weight row (output col) 0..127
      const int kk2 = i & 31;            // float2 index 0..31
      const float2 v = *(const float2*)(W + (size_t)j * Kdim + k0 + (kk2 << 1));
      *(float2*)&ldsW[j * LDSB_STRIDE + (kk2 << 1)] = v;
    }
    __syncthreads();

    if (ch + 1 < nchunks)                          // -> global_prefetch_b8
      __builtin_prefetch(Arow + k0 + KCHUNK, 0, 0);

    #pragma unroll
    for (int kk = 0; kk < KCHUNK; kk += 4) {
      // A fragment: 16x4 f32, lane holds A[row][k], A[row][k+1] (khalf-shifted)
      const float2 av = *(const float2*)(Arow + k0 + kk + khalf);
      v2f a; a.x = av.x; a.y = av.y;
      #pragma unroll
      for (int ct = 0; ct < 8; ++ct) {
        // B fragment: 4x16 f32, lane holds W[col][k], W[col][k+1]
        const float2 bv =
            *(const float2*)&ldsW[(ct * 16 + lrow) * LDSB_STRIDE + kk + khalf];
        v2f b; b.x = bv.x; b.y = bv.y;
        acc[ct] = __builtin_amdgcn_wmma_f32_16x16x4_f32(
            false, a, false, b, (short)0, acc[ct], false, false);
      }
    }
    __syncthreads();
  }

  if (active) {
    const int rhalf = (lane >> 4) << 3;  // lanes 16-31 hold M = r+8
    #pragma unroll
    for (int ct = 0; ct < 8; ++ct)
      #pragma unroll
      for (int r = 0; r < 8; ++r)
        C[(size_t)(rowBase + rhalf + r) * DD + ct * 16 + lrow] = acc[ct][r];
  }
}

// =====================================================================
// small helper kernels
// =====================================================================
__global__ void fill_kernel(float* p, float v, int n) {
  int i = blockIdx.x * blockDim.x + threadIdx.x;
  if (i < n) p[i] = v;
}

// el[n,h] = dot(ft[n,h,:], al[h,:]);  er likewise
__global__ void elr_kernel(const float* __restrict__ ft,
                           const float* __restrict__ al,
                           const float* __restrict__ ar,
                           float* __restrict__ el, float* __restrict__ er) {
  int idx = blockIdx.x * blockDim.x + threadIdx.x;   // n*H + h
  if (idx >= NN * HH) return;
  int n = idx >> 2, h = idx & 3;
  const float* f = ft + (size_t)n * DD + h * OUTF;
  const float* L = al + h * OUTF;
  const float* R = ar + h * OUTF;
  float sl = 0.f, sr = 0.f;
  #pragma unroll
  for (int i = 0; i < OUTF; ++i) { float v = f[i]; sl += v * L[i]; sr += v * R[i]; }
  el[idx] = sl; er[idx] = sr;
}

__device__ inline void atomicMaxF(float* addr, float val) {
  if (val >= 0.f) atomicMax((int*)addr, __float_as_int(val));
  else            atomicMin((unsigned int*)addr, __float_as_uint(val));
}

// e = leaky(el[src]+er[dst]); segment max into mx
__global__ void edge_e_kernel(const int* __restrict__ src, const int* __restrict__ dst,
                              const float* __restrict__ el, const float* __restrict__ er,
                              float* __restrict__ e, float* __restrict__ mx) {
  int idx = blockIdx.x * blockDim.x + threadIdx.x;
  if (idx >= EE * HH) return;
  int eid = idx >> 2, h = idx & 3;
  float v = el[src[eid] * HH + h] + er[dst[eid] * HH + h];
  v = v > 0.f ? v : NEG_SLOPE * v;
  e[idx] = v;
  atomicMaxF(&mx[dst[eid] * HH + h], v);
}

// ee = exp(e - mx[dst]); segment sum into sm   (in-place on e)
__global__ void edge_exp_kernel(const int* __restrict__ dst, float* __restrict__ e,
                                const float* __restrict__ mx, float* __restrict__ sm) {
  int idx = blockIdx.x * blockDim.x + threadIdx.x;
  if (idx >= EE * HH) return;
  int eid = idx >> 2, h = idx & 3;
  float v = __expf(e[idx] - mx[dst[eid] * HH + h]);
  e[idx] = v;
  atomicAdd(&sm[dst[eid] * HH + h], v);
}

// a = ee / sm[dst]   (in-place)
__global__ void edge_norm_kernel(const int* __restrict__ dst, float* __restrict__ e,
                                 const float* __restrict__ sm) {
  int idx = blockIdx.x * blockDim.x + threadIdx.x;
  if (idx >= EE * HH) return;
  int eid = idx >> 2, h = idx & 3;
  e[idx] = e[idx] / sm[dst[eid] * HH + h];
}

// global top-K per head via K iterative block argmax passes (buffer is L2-resident)
__global__ __launch_bounds__(256)
void topk_kernel(const float* __restrict__ a, float* __restrict__ tks,
                 int* __restrict__ tki) {
  const int h = blockIdx.x;      // head
  const int t = threadIdx.x;
  __shared__ float red_v[256];
  __shared__ int   red_i[256];
  __shared__ int   sel[TOPK];
  __shared__ float selv[TOPK];

  for (int k = 0; k < TOPK; ++k) {
    float best = -3.0e38f; int bi = -1;
    for (int e = t; e < EE; e += 256) {
      float v = a[(size_t)e * HH + h];
      bool used = false;
      for (int j = 0; j < k; ++j) used |= (sel[j] == e);
      if (!used && v > best) { best = v; bi = e; }
    }
    red_v[t] = best; red_i[t] = bi;
    __syncthreads();
    for (int s = 128; s > 0; s >>= 1) {
      if (t < s && red_v[t + s] > red_v[t]) { red_v[t] = red_v[t + s]; red_i[t] = red_i[t + s]; }
      __syncthreads();
    }
    if (t == 0) { sel[k] = red_i[0]; selv[k] = red_v[0]; }
    __syncthreads();
  }
  if (t == 0) {
    float m = selv[0];
    for (int k = 1; k < TOPK; ++k) m = fmaxf(m, selv[k]);
    float ex[TOPK]; float s = 0.f;
    for (int k = 0; k < TOPK; ++k) { ex[k] = __expf(selv[k] - m); s += ex[k]; }
    for (int k = 0; k < TOPK; ++k) { tks[h * TOPK + k] = ex[k] / s; tki[h * TOPK + k] = sel[k]; }
  }
}

// only K*H = 40 edges carry attention: sparse scatter of messages
__global__ void scatter_kernel(const int* __restrict__ src, const int* __restrict__ dst,
                               const float* __restrict__ ft,
                               const float* __restrict__ tks, const int* __restrict__ tki,
                               float* __restrict__ outb) {
  int b = blockIdx.x;                 // 0..H*TOPK-1
  int hh = b / TOPK, k = b % TOPK;
  int f = threadIdx.x;                // 0..31
  int e = tki[hh * TOPK + k];
  if (e < 0 || e >= EE) return;
  float s = tks[hh * TOPK + k];
  int sn = src[e], dn = dst[e];
  atomicAdd(&outb[(size_t)dn * DD + hh * OUTF + f],
            s * ft[(size_t)sn * DD + hh * OUTF + f]);
}

__global__ void elu_kernel(float* p, int n) {
  int i = blockIdx.x * blockDim.x + threadIdx.x;
  if (i >= n) return;
  float x = p[i];
  p[i] = x > 0.f ? x : (__expf(x) - 1.f);
}

__global__ void axpby_kernel(const float* __restrict__ a, const float* __restrict__ b,
                             float c0, float c1, float* __restrict__ o, int n) {
  int i = blockIdx.x * blockDim.x + threadIdx.x;
  if (i < n) o[i] = c0 * a[i] + c1 * b[i];
}

// w[n] = sum_j Wp2[j] * tanh(p[n,j] + bp1[j])
__global__ void sem_w_kernel(const float* __restrict__ p, const float* __restrict__ bp1,
                             const float* __restrict__ Wp2, float* __restrict__ w) {
  int n = blockIdx.x * blockDim.x + threadIdx.x;
  if (n >= NN) return;
  const float* pr = p + (size_t)n * DD;
  float acc = 0.f;
  for (int j = 0; j < DD; ++j) acc += Wp2[j] * tanhf(pr[j] + bp1[j]);
  w[n] = acc;
}

// beta = softmax(w0,w1) over M=2; out = beta0*emb0 + beta1*emb1
__global__ void combine_kernel(const float* __restrict__ emb0, const float* __restrict__ emb1,
                               const float* __restrict__ w0, const float* __restrict__ w1,
                               float* __restrict__ out) {
  int i = blockIdx.x * blockDim.x + threadIdx.x;
  if (i >= NN * DD) return;
  int n = i >> 7;
  float a = w0[n], b = w1[n];
  float m = fmaxf(a, b);
  float ea = __expf(a - m), eb = __expf(b - m);
  float inv = 1.f / (ea + eb);
  out[i] = (ea * emb0[i] + eb * emb1[i]) * inv;
}

// =====================================================================
// host orchestration
// =====================================================================
static void run_gat(const float* x, const int* src, const int* dst,
                    const float* fcW, const float* al, const float* ar,
                    float* ft, float* el, float* er, float* mx, float* sm,
                    float* ebuf, float* tks, int* tki, float* outb,
                    hipStream_t stream) {
  const int gemm_blocks = ((NN / 16) + 7) / 8;
  gemm_xwt_kernel<<<gemm_blocks, 256, 0, stream>>>(x, fcW, ft, NN, DD);
  elr_kernel<<<(NN * HH + 255) / 256, 256, 0, stream>>>(ft, al, ar, el, er);
  fill_kernel<<<(NN * HH + 255) / 256, 256, 0, stream>>>(mx, -3.0e38f, NN * HH);
  fill_kernel<<<(NN * HH + 255) / 256, 256, 0, stream>>>(sm, 0.f, NN * HH);
  edge_e_kernel<<<(EE * HH + 255) / 256, 256, 0, stream>>>(src, dst, el, er, ebuf, mx);
  edge_exp_kernel<<<(EE * HH + 255) / 256, 256, 0, stream>>>(dst, ebuf, mx, sm);
  edge_norm_kernel<<<(EE * HH + 255) / 256, 256, 0, stream>>>(dst, ebuf, sm);
  topk_kernel<<<HH, 256, 0, stream>>>(ebuf, tks, tki);
  fill_kernel<<<(NN * DD + 255) / 256, 256, 0, stream>>>(outb, 0.f, NN * DD);
  scatter_kernel<<<HH * TOPK, 32, 0, stream>>>(src, dst, ft, tks, tki, outb);
  elu_kernel<<<(NN * DD + 255) / 256, 256, 0, stream>>>(outb, NN * DD);
}

extern "C" void kernel_launch(void* const* d_in, const int* in_sizes, int n_in,
                              void* d_out, int out_size, void* d_ws, size_t ws_size,
                              hipStream_t stream) {
  const float* h    = (const float*)d_in[0];
  const int*   src0 = (const int*)  d_in[1];
  const int*   dst0 = (const int*)  d_in[2];
  const int*   src1 = (const int*)  d_in[3];
  const int*   dst1 = (const int*)  d_in[4];
  const float* Wh   = (const float*)d_in[5];
  const float* fc0  = (const float*)d_in[6];
  const float* al0  = (const float*)d_in[7];
  const float* ar0  = (const float*)d_in[8];
  const float* fc1  = (const float*)d_in[9];
  const float* al1  = (const float*)d_in[10];
  const float* ar1  = (const float*)d_in[11];
  const float* Wp1  = (const float*)d_in[12];
  const float* bp1  = (const float*)d_in[13];
  const float* Wp2  = (const float*)d_in[14];
  float* out = (float*)d_out;

  // workspace carve-up (floats)
  float* ws = (float*)d_ws;
  size_t off = 0;
  auto alloc = [&](size_t n) { float* p = ws + off; off += n; return p; };
  float* wh   = alloc((size_t)NN * DD);
  float* ft   = alloc((size_t)NN * DD);   // per-layer projection; reused as pbuf
  float* aout = alloc((size_t)NN * DD);
  float* bout = alloc((size_t)NN * DD);
  float* emb0 = alloc((size_t)NN * DD);
  float* emb1 = alloc((size_t)NN * DD);
  float* el   = alloc((size_t)NN * HH);
  float* er   = alloc((size_t)NN * HH);
  float* mx   = alloc((size_t)NN * HH);
  float* sm   = alloc((size_t)NN * HH);
  float* ebuf = alloc((size_t)EE * HH);
  float* w0   = alloc((size_t)NN);
  float* w1   = alloc((size_t)NN);
  float* tks  = alloc((size_t)HH * TOPK);
  int*   tki  = (int*)alloc((size_t)HH * TOPK);

  // decay(2, 1.5): c0 = 1/(1 + 2^-1.5), c1 = 1 - c0
  const float x1 = 1.0f / (2.0f * sqrtf(2.0f));
  const float c0 = 1.0f / (1.0f + x1);
  const float c1 = x1 / (1.0f + x1);

  const int gemm_blocks = ((NN / 16) + 7) / 8;

  // wh = h @ Wh.T  (K = 256)
  gemm_xwt_kernel<<<gemm_blocks, 256, 0, stream>>>(h, Wh, wh, NN, INF_);

  const int* srcs[MM] = { src0, src1 };
  const int* dsts[MM] = { dst0, dst1 };
  float* embs[MM] = { emb0, emb1 };
  for (int m = 0; m < MM; ++m) {
    run_gat(wh, srcs[m], dsts[m], fc0 + (size_t)m * DD * WSZ,
            al0 + (size_t)m * HH * OUTF, ar0 + (size_t)m * HH * OUTF,
            ft, el, er, mx, sm, ebuf, tks, tki, aout, stream);
    run_gat(aout, srcs[m], dsts[m], fc1 + (size_t)m * DD * DD,
            al1 + (size_t)m * HH * OUTF, ar1 + (size_t)m * HH * OUTF,
            ft, el, er, mx, sm, ebuf, tks, tki, bout, stream);
    axpby_kernel<<<(NN * DD + 255) / 256, 256, 0, stream>>>(aout, bout, c0, c1,
                                                            embs[m], NN * DD);
  }

  // semantic attention
  gemm_xwt_kernel<<<gemm_blocks, 256, 0, stream>>>(emb0, Wp1, ft, NN, DD);
  sem_w_kernel<<<(NN + 255) / 256, 256, 0, stream>>>(ft, bp1, Wp2, w0);
  gemm_xwt_kernel<<<gemm_blocks, 256, 0, stream>>>(emb1, Wp1, ft, NN, DD);
  sem_w_kernel<<<(NN + 255) / 256, 256, 0, stream>>>(ft, bp1, Wp2, w1);
  combine_kernel<<<(NN * DD + 255) / 256, 256, 0, stream>>>(emb0, emb1, w0, w1, out);
}